// GCN_36412732735562
// MI455X (gfx1250) — compile-verified
//
#include <hip/hip_runtime.h>
#include <hip/hip_bf16.h>

typedef __attribute__((ext_vector_type(2))) float v2f;
typedef __attribute__((ext_vector_type(8))) float v8f;

// Global-address-space pointer helpers: integer round-trip survives loops/phis
// so loads/stores/atomics lower to global_* (never flat_*).
typedef __attribute__((address_space(1))) float gfloat;
typedef __attribute__((address_space(1))) const float gcfloat;
static __device__ __forceinline__ gcfloat* gci(const float* p) {
  return (gcfloat*)(unsigned long long)p;
}
static __device__ __forceinline__ gfloat* gmi(float* p) {
  return (gfloat*)(unsigned long long)p;
}

#define LEAKY_SLOPE 0.2f

// ---------------- degree kernels ----------------
__global__ void k_deg_init(float* __restrict__ deg, int N) {
  int i = blockIdx.x * blockDim.x + threadIdx.x;
  if (i < N) gmi(deg)[i] = 1.0f;                  // self-loop contribution
}

__global__ void k_deg_count(const int* __restrict__ dst, float* __restrict__ deg, int E) {
  int i = blockIdx.x * blockDim.x + threadIdx.x;
  if (i < E)
    __hip_atomic_fetch_add(gmi(deg) + dst[i], 1.0f,
                           __ATOMIC_RELAXED, __HIP_MEMORY_SCOPE_AGENT);
}

__global__ void k_deg_fin(float* __restrict__ deg, int N) {
  int i = blockIdx.x * blockDim.x + threadIdx.x;
  if (i < N) gmi(deg)[i] = 1.0f / sqrtf(gci(deg)[i]);   // deg >= 1 always
}

// ---------------- bias broadcast init ----------------
__global__ void k_fill_bias(float* __restrict__ out, const float* __restrict__ bias,
                            int N, int F) {
  int i = blockIdx.x * blockDim.x + threadIdx.x;
  if (i < N * F) gmi(out)[i] = gci(bias)[i % F];
}

// ---------------- WMMA fp32 GEMM: C[M,Nc] = act(A[M,K]) x B[K,Nc] ----------------
// Each wave computes NT adjacent 16x16 output tiles (shared A fragment, NT
// independent WMMA accumulator chains), K consumed 4/step via
// V_WMMA_F32_16X16X4_F32. Pure 32-bit integer offsets: uniform K-offset in
// SALU, per-lane column offsets loop-invariant -> saddr-form global loads.
// Loads for the whole step are gathered before the NT WMMAs (waits stagger).
// GUARD=true handles Nc % 16 != 0 (clamped column + 0/1 mask), else free.
template <int NT, bool APPLY_LEAKY, bool GUARD>
__global__ void k_gemm(const float* __restrict__ A, const float* __restrict__ B,
                       float* __restrict__ C, int M, int K, int Nc) {
  const int tilesM  = (M + 15) >> 4;
  const int tilesN  = (Nc + 15) >> 4;
  const int tilesNg = (tilesN + NT - 1) / NT;     // tile-groups along N
  const int wave = blockIdx.x * (blockDim.x >> 5) + (threadIdx.x >> 5);
  const int tM  = wave / tilesNg;
  const int tNg = wave - tM * tilesNg;
  if (tM >= tilesM) return;                       // wave-uniform exit

  const int lane = threadIdx.x & 31;
  const int half = lane >> 4;                     // 0: K=0..1, 1: K=2..3 (ISA A/B layout)
  const int l16  = lane & 15;

  gcfloat* __restrict__ Ab = gci(A);
  gcfloat* __restrict__ Bb = gci(B);

  int rowA = tM * 16 + l16;
  if (rowA >= M) rowA = M - 1;                    // stores are guarded
  int aoff = rowA * K + half * 2;                 // fits in 32 bits for these sizes

  int   col [NT];                                 // true output column
  int   bcol[NT];                                 // clamped, always in-bounds
  float bmul[NT];                                 // 1/0 mask (GUARD only)
#pragma unroll
  for (int j = 0; j < NT; ++j) {
    int c = (tNg * NT + j) * 16 + l16;
    col[j]  = c;
    bcol[j] = c;
    if (GUARD) {
      bmul[j] = (c < Nc) ? 1.0f : 0.0f;
      bcol[j] = (c < Nc) ? c : (Nc - 1);
    }
  }

  v8f acc[NT];
#pragma unroll
  for (int j = 0; j < NT; ++j) acc[j] = {};

  int boff = half * 2 * Nc;                       // per-lane: half selects B rows 2*half, 2*half+1
  for (int k = 0; k < K; k += 4) {
    // ---- gather all fragments for this K-step ----
    float a0 = Ab[aoff];
    float a1 = Ab[aoff + 1];
    if (APPLY_LEAKY) {
      a0 = (a0 > 0.0f) ? a0 : LEAKY_SLOPE * a0;
      a1 = (a1 > 0.0f) ? a1 : LEAKY_SLOPE * a1;
    }
    const v2f av = {a0, a1};
    v2f bv[NT];
#pragma unroll
    for (int j = 0; j < NT; ++j) {
      float b0 = Bb[boff + bcol[j]];
      float b1 = Bb[boff + Nc + bcol[j]];
      if (GUARD) { b0 *= bmul[j]; b1 *= bmul[j]; }
      bv[j] = {b0, b1};
    }
    // ---- NT independent accumulator chains ----
#pragma unroll
    for (int j = 0; j < NT; ++j)
      acc[j] = __builtin_amdgcn_wmma_f32_16x16x4_f32(false, av, false, bv[j],
                                                     (short)0, acc[j], false, false);
    aoff += 4;
    boff += 4 * Nc;
  }

  gfloat* __restrict__ Cb = gmi(C);
  const int row0 = tM * 16 + half * 8;            // C/D layout: lanes16-31 hold M=8..15
#pragma unroll
  for (int j = 0; j < NT; ++j) {
    if (col[j] < Nc) {
#pragma unroll
      for (int r = 0; r < 8; ++r) {
        const int row = row0 + r;
        if (row < M) Cb[(long long)row * Nc + col[j]] = acc[j][r];
      }
    }
  }
}

// ---------------- edge scatter: out[dst] += H[src] * dinv[src]*dinv[dst] ----------------
// TPE threads per edge, each moves F/TPE consecutive floats (coalesced).
// Edges e >= E are the N self-loops (src = dst = e - E).
template <int F, int TPE>
__global__ void k_scatter(const float* __restrict__ H, const int* __restrict__ ei,
                          const float* __restrict__ dinv, float* __restrict__ out,
                          int E, int N) {
  constexpr int CHUNK = F / TPE;
  const long long idx   = (long long)blockIdx.x * blockDim.x + threadIdx.x;
  const long long total = (long long)(E + N) * TPE;
  if (idx >= total) return;
  const long long e = idx / TPE;
  const int part    = (int)(idx - e * TPE);

  int src, dst;
  if (e < (long long)E) {
    src = ei[e];
    dst = ei[(long long)E + e];
  } else {
    src = dst = (int)(e - E);
  }
  const float norm = gci(dinv)[src] * gci(dinv)[dst];

  gcfloat* __restrict__ hs = gci(H)   + (long long)src * F + part * CHUNK;
  gfloat*  __restrict__ od = gmi(out) + (long long)dst * F + part * CHUNK;
  float v[CHUNK];
#pragma unroll
  for (int c = 0; c < CHUNK; ++c) v[c] = hs[c];   // merged into wide global loads
#pragma unroll
  for (int c = 0; c < CHUNK; ++c)
    __hip_atomic_fetch_add(od + c, v[c] * norm,
                           __ATOMIC_RELAXED, __HIP_MEMORY_SCOPE_AGENT);
}

// ---------------- host side ----------------
static inline int cdiv_i(long long a, long long b) { return (int)((a + b - 1) / b); }

extern "C" void kernel_launch(void* const* d_in, const int* in_sizes, int n_in,
                              void* d_out, int out_size, void* d_ws, size_t ws_size,
                              hipStream_t stream) {
  const float* x  = (const float*)d_in[0];
  const int*   ei = (const int*)  d_in[1];     // [2,E] int32: row 0 = src, row 1 = dst
  const float* W1 = (const float*)d_in[2];
  const float* b1 = (const float*)d_in[3];
  const float* W2 = (const float*)d_in[4];
  const float* b2 = (const float*)d_in[5];
  const float* W3 = (const float*)d_in[6];
  const float* b3 = (const float*)d_in[7];
  float* out = (float*)d_out;

  const int F1 = in_sizes[3];                  // 128
  const int F2 = in_sizes[5];                  // 16
  const int Cc = in_sizes[7];                  // 40
  const int F0 = in_sizes[2] / F1;             // 256
  const int N  = in_sizes[0] / F0;             // 100000
  const int E  = in_sizes[1] / 2;              // 3200000

  // workspace layout (floats): dinv[N] | H1[N*F1] | H1agg[N*F1] | H2agg[N*F2]
  float* ws    = (float*)d_ws;
  float* dinv  = ws;
  float* H1    = dinv  + N;
  float* H1agg = H1    + (size_t)N * F1;
  float* H2agg = H1agg + (size_t)N * F1;
  float* H2    = H1;                            // reuse: H1 dead after scatter-1
  float* H3    = H1agg;                         // reuse: H1agg dead after GEMM-2 (Cc <= F1)

  const int TB = 256;

  // degree / normalization
  k_deg_init <<<cdiv_i(N, TB), TB, 0, stream>>>(dinv, N);
  k_deg_count<<<cdiv_i(E, TB), TB, 0, stream>>>(ei + E, dinv, E);
  k_deg_fin  <<<cdiv_i(N, TB), TB, 0, stream>>>(dinv, N);

  const int tilesM = (N + 15) / 16;

  // layer 1: H1 = x @ W1 ; H1agg = b1 + scatter(H1)      (NT=4 -> 2 wave-groups/row)
  {
    long long waves = (long long)tilesM * ((((F1 + 15) / 16) + 3) / 4);
    k_gemm<4, false, false><<<cdiv_i(waves, 8), TB, 0, stream>>>(x, W1, H1, N, F0, F1);
  }
  k_fill_bias<<<cdiv_i((long long)N * F1, TB), TB, 0, stream>>>(H1agg, b1, N, F1);
  k_scatter<128, 32><<<cdiv_i((long long)(E + N) * 32, TB), TB, 0, stream>>>(
      H1, ei, dinv, H1agg, E, N);

  // layer 2: H2 = leaky(H1agg) @ W2 ; H2agg = b2 + scatter(H2)   (NT=1, 16 cols)
  {
    long long waves = (long long)tilesM * ((F2 + 15) / 16);
    k_gemm<1, true, false><<<cdiv_i(waves, 8), TB, 0, stream>>>(H1agg, W2, H2, N, F1, F2);
  }
  k_fill_bias<<<cdiv_i((long long)N * F2, TB), TB, 0, stream>>>(H2agg, b2, N, F2);
  k_scatter<16, 4><<<cdiv_i((long long)(E + N) * 4, TB), TB, 0, stream>>>(
      H2, ei, dinv, H2agg, E, N);

  // layer 3: H3 = leaky(H2agg) @ W3 ; out = b3 + scatter(H3)   (NT=3 covers 48 cols)
  {
    long long waves = (long long)tilesM * 1;    // 3 tiles / NT=3 -> one group
    k_gemm<3, true, true><<<cdiv_i(waves, 8), TB, 0, stream>>>(H2agg, W3, H3, N, F2, Cc);
  }
  k_fill_bias<<<cdiv_i((long long)N * Cc, TB), TB, 0, stream>>>(out, b3, N, Cc);
  k_scatter<40, 8><<<cdiv_i((long long)(E + N) * 8, TB), TB, 0, stream>>>(
      H3, ei, dinv, out, E, N);
}